// UserFusion_70892730188381
// MI455X (gfx1250) — compile-verified
//
#include <hip/hip_runtime.h>
#include <math.h>

#define EMB 64
#define EPSR 1e-8f

typedef __attribute__((ext_vector_type(2))) float v2f;
typedef __attribute__((ext_vector_type(8))) float v8f;

// ---- degree count: cnt[idx[e]] += 1.0f ------------------------------------
__global__ void uf_count(const int* __restrict__ idx, float* __restrict__ cnt, int n) {
  int e = blockIdx.x * blockDim.x + threadIdx.x;
  if (e < n) atomicAdd(&cnt[idx[e]], 1.0f);
}

// ---- edge row scatter: acc[dst[e]] += tab[src[e]]  (float4 per thread) ----
__global__ void uf_scatter(const float* __restrict__ tab, const int* __restrict__ sidx,
                           const int* __restrict__ didx, float* __restrict__ acc,
                           int n_edges) {
  int t = blockIdx.x * blockDim.x + threadIdx.x;
  int e = t >> 4;
  int c = (t & 15) << 2;
  if (e >= n_edges) return;
  int s = sidx[e];
  int d = didx[e];
  const float4 v = *(const float4*)(tab + (size_t)s * EMB + c);
  float* p = acc + (size_t)d * EMB + c;
  atomicAdd(p + 0, v.x);
  atomicAdd(p + 1, v.y);
  atomicAdd(p + 2, v.z);
  atomicAdd(p + 3, v.w);
}

// ---- in-place segment mean: acc[r] /= max(cnt[r],1) -----------------------
__global__ void uf_mean(float* __restrict__ acc, const float* __restrict__ cnt, int rows) {
  int t = blockIdx.x * blockDim.x + threadIdx.x;
  int r = t >> 4;
  int c = (t & 15) << 2;
  if (r >= rows) return;
  float inv = 1.0f / fmaxf(cnt[r], 1.0f);
  float4* p = (float4*)(acc + (size_t)r * EMB + c);
  float4 v = *p;
  v.x *= inv; v.y *= inv; v.z *= inv; v.w *= inv;
  *p = v;
}

// ---- in-place rating conv: acc[r] = sw(r)*self[r] + acc[r]/max(cnt,1) -----
__global__ void uf_rat(float* __restrict__ acc, const float* __restrict__ self,
                       const float* __restrict__ cnt, int rows) {
  int t = blockIdx.x * blockDim.x + threadIdx.x;
  int r = t >> 4;
  int c = (t & 15) << 2;
  if (r >= rows) return;
  float cr = cnt[r];
  float inv = 1.0f / fmaxf(cr, 1.0f);
  float sw = 1.0f - cr / (cr + EPSR);   // matches reference exactly
  float4* p = (float4*)(acc + (size_t)r * EMB + c);
  float4 a = *p;
  const float4 s = *(const float4*)(self + (size_t)r * EMB + c);
  a.x = sw * s.x + a.x * inv;
  a.y = sw * s.y + a.y * inv;
  a.z = sw * s.z + a.z * inv;
  a.w = sw * s.w + a.w * inv;
  *p = a;
}

// ---- batch gather + fused layer-2 finish ----------------------------------
// final_user = ue + 0.5*(soc1 + ratu1) + 0.5*(soc2acc/cs + usw*ratu1 + like2acc/cu)
__global__ void uf_gather(const float* __restrict__ user_emb, const float* __restrict__ item_emb,
                          const int* __restrict__ users, const int* __restrict__ items,
                          const float* __restrict__ cnt_soc, const float* __restrict__ cnt_u,
                          const float* __restrict__ A1,  // soc1
                          const float* __restrict__ A2,  // rat_u1
                          const float* __restrict__ A4,  // soc2 accum (unnormalized)
                          const float* __restrict__ A5,  // like2 accum (unnormalized)
                          float* __restrict__ out_lu, float* __restrict__ out_li,
                          int batch) {
  int t = blockIdx.x * blockDim.x + threadIdx.x;
  int b = t >> 4;
  int c = (t & 15) << 2;
  if (b >= batch) return;
  int u = users[b];
  int it = items[b];
  float cs_inv = 1.0f / fmaxf(cnt_soc[u], 1.0f);
  float cur = cnt_u[u];
  float cu_inv = 1.0f / fmaxf(cur, 1.0f);
  float usw = 1.0f - cur / (cur + EPSR);
  size_t ro = (size_t)u * EMB + c;
  const float4 ue = *(const float4*)(user_emb + ro);
  const float4 s1 = *(const float4*)(A1 + ro);
  const float4 r1 = *(const float4*)(A2 + ro);
  const float4 s2 = *(const float4*)(A4 + ro);
  const float4 l2 = *(const float4*)(A5 + ro);
  float4 fu;
  fu.x = ue.x + 0.5f * (s1.x + r1.x) + 0.5f * (s2.x * cs_inv + usw * r1.x + l2.x * cu_inv);
  fu.y = ue.y + 0.5f * (s1.y + r1.y) + 0.5f * (s2.y * cs_inv + usw * r1.y + l2.y * cu_inv);
  fu.z = ue.z + 0.5f * (s1.z + r1.z) + 0.5f * (s2.z * cs_inv + usw * r1.z + l2.z * cu_inv);
  fu.w = ue.w + 0.5f * (s1.w + r1.w) + 0.5f * (s2.w * cs_inv + usw * r1.w + l2.w * cu_inv);
  *(float4*)(out_lu + (size_t)b * EMB + c) = fu;
  const float4 ie = *(const float4*)(item_emb + (size_t)it * EMB + c);
  *(float4*)(out_li + (size_t)b * EMB + c) = ie;
}

// ---- predict = sigmoid(diag(U * I^T)) via V_WMMA_F32_16X16X4_F32 ----------
// One wave = one 16-row tile. A lane layout (ISA 7.12.2, f32 16x4 A):
//   lanes 0-15 hold row M=lane, K = {0,1}; lanes 16-31 hold row M=lane-16, K = {2,3}.
// B (4x16) mirrors it with lanes = columns N (item rows). K advances 4 per WMMA, x16.
// C/D diag: D[m][m] -> c[m] at lane m (m<8); c[m-8] at lane m+16 (m>=8).
__global__ __launch_bounds__(256) void uf_predict(const float* __restrict__ lu,
                                                  const float* __restrict__ li,
                                                  float* __restrict__ pred, int batch) {
  int wave = blockIdx.x * (blockDim.x >> 5) + (threadIdx.x >> 5);
  int base = wave * 16;
  if (base >= batch) return;               // wave-uniform: EXEC stays all-ones for WMMA
  int lane = threadIdx.x & 31;
  int r = lane & 15;
  int row = base + r;
  if (row > batch - 1) row = batch - 1;
  int koff = (lane < 16) ? 0 : 2;
  const float* up = lu + (size_t)row * EMB + koff;
  const float* ip = li + (size_t)row * EMB + koff;
  v8f c = {};
#pragma unroll
  for (int k = 0; k < EMB; k += 4) {
    v2f a, b;
    a.x = up[k];     a.y = up[k + 1];
    b.x = ip[k];     b.y = ip[k + 1];
    c = __builtin_amdgcn_wmma_f32_16x16x4_f32(false, a, false, b, (short)0, c, false, false);
  }
  // extract this lane's diagonal element
  int sel = lane & 7;
  float d = c[0];
  if (sel == 1) d = c[1];
  else if (sel == 2) d = c[2];
  else if (sel == 3) d = c[3];
  else if (sel == 4) d = c[4];
  else if (sel == 5) d = c[5];
  else if (sel == 6) d = c[6];
  else if (sel == 7) d = c[7];
  int m = -1;
  if (lane < 8) m = lane;                  // c[m] @ lane m  -> D[m][m]
  else if (lane >= 24) m = lane - 16;      // c[lane-24] @ lane -> D[lane-16][lane-16]
  if (m >= 0) {
    int orow = base + m;
    if (orow < batch) pred[orow] = 1.0f / (1.0f + expf(-d));
  }
}

extern "C" void kernel_launch(void* const* d_in, const int* in_sizes, int n_in,
                              void* d_out, int out_size, void* d_ws, size_t ws_size,
                              hipStream_t stream) {
  (void)n_in; (void)out_size; (void)ws_size;
  const float* user_emb = (const float*)d_in[0];
  const float* item_emb = (const float*)d_in[1];
  const int* users   = (const int*)d_in[2];
  const int* items   = (const int*)d_in[3];
  const int* soc_src = (const int*)d_in[4];
  const int* soc_dst = (const int*)d_in[5];
  const int* like_u  = (const int*)d_in[6];
  const int* like_i  = (const int*)d_in[7];

  const int num_users = in_sizes[0] / EMB;
  const int num_items = in_sizes[1] / EMB;
  const int batch = in_sizes[2];
  const int n_soc = in_sizes[4];
  const int n_rat = in_sizes[6];

  float* ws = (float*)d_ws;
  size_t o = 0;
  float* cnt_soc = ws + o; o += (size_t)num_users;
  float* cnt_u   = ws + o; o += (size_t)num_users;
  float* cnt_i   = ws + o; o += (size_t)num_items;
  float* A1 = ws + o; o += (size_t)num_users * EMB;   // social accum -> soc1
  float* A2 = ws + o; o += (size_t)num_users * EMB;   // like accum   -> rat_u1
  float* A3 = ws + o; o += (size_t)num_items * EMB;   // rev accum    -> rat_i1
  float* A4 = ws + o; o += (size_t)num_users * EMB;   // social layer-2 accum
  float* A5 = ws + o; o += (size_t)num_users * EMB;   // like layer-2 accum
  hipMemsetAsync(d_ws, 0, o * sizeof(float), stream);

  float* pred   = (float*)d_out;
  float* out_lu = pred + batch;
  float* out_li = out_lu + (size_t)batch * EMB;

  const int B = 256;
  // degrees
  uf_count<<<(n_soc + B - 1) / B, B, 0, stream>>>(soc_dst, cnt_soc, n_soc);
  uf_count<<<(n_rat + B - 1) / B, B, 0, stream>>>(like_u, cnt_u, n_rat);
  uf_count<<<(n_rat + B - 1) / B, B, 0, stream>>>(like_i, cnt_i, n_rat);
  // layer-1 scatters
  long ts = (long)n_soc * 16, tr = (long)n_rat * 16;
  uf_scatter<<<(ts + B - 1) / B, B, 0, stream>>>(user_emb, soc_src, soc_dst, A1, n_soc);
  uf_scatter<<<(tr + B - 1) / B, B, 0, stream>>>(item_emb, like_i, like_u, A2, n_rat);
  uf_scatter<<<(tr + B - 1) / B, B, 0, stream>>>(user_emb, like_u, like_i, A3, n_rat);
  // normalize layer-1 (in place)
  long tu = (long)num_users * 16, ti = (long)num_items * 16;
  uf_mean<<<(tu + B - 1) / B, B, 0, stream>>>(A1, cnt_soc, num_users);
  uf_rat <<<(tu + B - 1) / B, B, 0, stream>>>(A2, user_emb, cnt_u, num_users);
  uf_rat <<<(ti + B - 1) / B, B, 0, stream>>>(A3, item_emb, cnt_i, num_items);
  // layer-2 scatters
  uf_scatter<<<(ts + B - 1) / B, B, 0, stream>>>(A1, soc_src, soc_dst, A4, n_soc);
  uf_scatter<<<(tr + B - 1) / B, B, 0, stream>>>(A3, like_i, like_u, A5, n_rat);
  // batch gather + final-user fuse
  long tb = (long)batch * 16;
  uf_gather<<<(tb + B - 1) / B, B, 0, stream>>>(user_emb, item_emb, users, items,
                                                cnt_soc, cnt_u, A1, A2, A4, A5,
                                                out_lu, out_li, batch);
  // predict via f32 WMMA
  int tiles = (batch + 15) / 16;
  int wpb = 8;  // 8 waves (256 threads) per block
  uf_predict<<<(tiles + wpb - 1) / wpb, wpb * 32, 0, stream>>>(out_lu, out_li, pred, batch);
}